// Separator_86517821213140
// MI455X (gfx1250) — compile-verified
//
#include <hip/hip_runtime.h>
#include <math.h>

typedef unsigned int u32;
typedef __attribute__((ext_vector_type(4))) u32   v4u;
typedef __attribute__((ext_vector_type(8))) int   v8i;
typedef __attribute__((ext_vector_type(4))) int   v4i;
typedef __attribute__((ext_vector_type(2))) float v2f;
typedef __attribute__((ext_vector_type(8))) float v8f;

#define NF    512      // frames
#define NB    2049     // bins
#define YROW  16       // floats per (frame,bin) in y layout: (c,ri,s) = 2*2*4
#define XROW  4        // floats per (frame,bin) in x layout: (c,ri)
#define NS    4        // sources
#define EPSV  1e-10f

// ---- Tensor Data Mover: 2D tile load Global -> LDS (D# per CDNA5 ISA §8.3-8.5)
__device__ __forceinline__ void tdm_load_2d(u32 lds_addr, const void* gptr,
                                            u32 tile_w, u32 rows, u32 stride_elems) {
  unsigned long long ga = (unsigned long long)(uintptr_t)gptr;
  v4u g0;
  g0[0] = 1u;                                            // count=1 (valid), no gather
  g0[1] = lds_addr;                                      // LDS byte address
  g0[2] = (u32)(ga & 0xFFFFFFFFull);                     // global_addr[31:0]
  g0[3] = (u32)((ga >> 32) & 0x01FFFFFFull) | (2u << 30);// global_addr[56:32] | type=2
  v8i g1;
  g1[0] = (int)(2u << 16);          // workgroup_mask=0, data_size=2 (4B)
  g1[1] = (int)(tile_w << 16);      // tensor_dim0 low16 (bits 63:48)
  g1[2] = (int)(rows   << 16);      // tensor_dim0 hi=0 ; tensor_dim1 low16
  g1[3] = (int)(tile_w << 16);      // tensor_dim1 hi=0 ; tile_dim0
  g1[4] = (int)rows;                // tile_dim1 ; tile_dim2=0
  g1[5] = (int)stride_elems;        // tensor_dim0_stride low32
  g1[6] = 0;                        // stride hi ; tensor_dim1_stride low
  g1[7] = 0;
  v4i z4 = {0, 0, 0, 0};
  v8i z8 = {0, 0, 0, 0, 0, 0, 0, 0};
  __builtin_amdgcn_tensor_load_to_lds(g0, g1, z4, z4, z8, 0);
}

// G (16x16 f32 WMMA D-matrix) flattened as [vgpr*32 + lane]:
// element (m,n): vgpr = m&7, lane = n + ((m>=8)?16:0)
#define GE(Garr, m, n) Garr[(((m) & 7) << 5) + (n) + (((m) >> 3) << 4)]

__global__ void __launch_bounds__(256)
em_wiener_kernel(const float* __restrict__ y, const float* __restrict__ x,
                 const int* __restrict__ iters, float* __restrict__ out) {
  __shared__ __align__(16) float Ysh[NF * YROW];   // 32 KB: per-bin y tile (in-place updated)
  __shared__ __align__(16) float Xsh[NF * XROW];   //  8 KB: per-bin x tile
  __shared__ float Gpart[8 * 256];                 //  8 KB: per-wave Gram partials
  __shared__ float G[256];                         //  1 KB: reduced 16x16 Gram
  __shared__ float Rsh[NS][2][2][2];               // [s][i][j][re/im]

  const int tid   = threadIdx.x;
  const int b     = blockIdx.x;
  const int niter = iters[0];

  // ---------------- bulk strided tile loads via TDM (wave 0 issues, all wait) ---------
  if (tid < 32) {
    tdm_load_2d((u32)(uintptr_t)Ysh, y + (size_t)b * YROW, YROW, NF, NB * YROW);
    tdm_load_2d((u32)(uintptr_t)Xsh, x + (size_t)b * XROW, XROW, NF, NB * XROW);
    __builtin_amdgcn_s_wait_tensorcnt(0);
  }
  __syncthreads();

  // Z column mapping: col m = s*4 + t, t -> (c=t&1, ri=t>>1); raw offset = c*8 + ri*4 + s
  const int lane = tid & 31;
  const int wv   = tid >> 5;
  const int half = lane >> 4;        // K-pair selector for 16x4 A layout
  const int mcol = lane & 15;
  const int sc   = mcol >> 2, tc = mcol & 3;
  const int zoff = ((tc & 1) << 3) + ((tc >> 1) << 2) + sc;

  const float reg = sqrtf(EPSV);

  for (int it = 0; it < niter; ++it) {
    // ---- Gram G = Z^T Z (16x16, K=512) via V_WMMA_F32_16X16X4_F32; 64 frames / wave
    v8f acc = {0.f, 0.f, 0.f, 0.f, 0.f, 0.f, 0.f, 0.f};
    const int fbase = wv * 64;
    #pragma unroll
    for (int ch = 0; ch < 16; ++ch) {
      const int f0 = fbase + ch * 4 + 2 * half;   // A 16x4: v0={K0,K2}, v1={K1,K3}
      v2f aop;
      aop.x = Ysh[f0 * YROW + zoff];
      aop.y = Ysh[(f0 + 1) * YROW + zoff];
      acc = __builtin_amdgcn_wmma_f32_16x16x4_f32(false, aop, false, aop,
                                                  (short)0, acc, false, false);
    }
    #pragma unroll
    for (int r = 0; r < 8; ++r) Gpart[wv * 256 + r * 32 + lane] = acc[r];
    __syncthreads();

    // ---- reduce 8 wave partials
    {
      float g = 0.f;
      #pragma unroll
      for (int w2 = 0; w2 < 8; ++w2) g += Gpart[w2 * 256 + tid];
      G[tid] = g;
    }
    __syncthreads();

    // ---- spatial covariance R[s] = num / den   (threads 0..3, one source each)
    if (tid < NS) {
      const int s4 = tid * 4;
      const float den = EPSV + 0.5f * (GE(G, s4, s4) + GE(G, s4 + 1, s4 + 1) +
                                       GE(G, s4 + 2, s4 + 2) + GE(G, s4 + 3, s4 + 3));
      const float id = 1.f / den;
      #pragma unroll
      for (int i = 0; i < 2; ++i)
        #pragma unroll
        for (int j = 0; j < 2; ++j) {
          const float nre = GE(G, s4 + i, s4 + j) + GE(G, s4 + 2 + i, s4 + 2 + j);
          const float nim = GE(G, s4 + 2 + i, s4 + j) - GE(G, s4 + i, s4 + 2 + j);
          Rsh[tid][i][j][0] = nre * id;
          Rsh[tid][i][j][1] = nim * id;
        }
    }
    __syncthreads();

    // ---- per-frame Wiener update (2 frames per thread), in-place in Ysh
    #pragma unroll
    for (int rep = 0; rep < 2; ++rep) {
      const int f = tid + rep * 256;
      float* yf = &Ysh[f * YROW];
      float vs[NS];
      #pragma unroll
      for (int s = 0; s < NS; ++s) {
        const float a0 = yf[s], a1 = yf[4 + s], a2 = yf[8 + s], a3 = yf[12 + s];
        vs[s] = 0.5f * (a0 * a0 + a1 * a1 + a2 * a2 + a3 * a3);
      }
      // Cxx = reg*I + sum_s vs[s] * R[s]
      float c00r = reg, c00i = 0.f, c01r = 0.f, c01i = 0.f;
      float c10r = 0.f, c10i = 0.f, c11r = reg, c11i = 0.f;
      #pragma unroll
      for (int s = 0; s < NS; ++s) {
        c00r += vs[s] * Rsh[s][0][0][0]; c00i += vs[s] * Rsh[s][0][0][1];
        c01r += vs[s] * Rsh[s][0][1][0]; c01i += vs[s] * Rsh[s][0][1][1];
        c10r += vs[s] * Rsh[s][1][0][0]; c10i += vs[s] * Rsh[s][1][0][1];
        c11r += vs[s] * Rsh[s][1][1][0]; c11i += vs[s] * Rsh[s][1][1][1];
      }
      // 2x2 complex inverse, inv_det = conj(det)/|det|^2
      const float dre = (c00r * c11r - c00i * c11i) - (c01r * c10r - c01i * c10i);
      const float dim = (c00r * c11i + c00i * c11r) - (c01r * c10i + c01i * c10r);
      const float dn  = dre * dre + dim * dim;
      const float ir  =  dre / dn;
      const float ii  = -dim / dn;
      const float i00r =  c11r * ir - c11i * ii, i00i =  c11r * ii + c11i * ir;
      const float i01r = -(c01r * ir - c01i * ii), i01i = -(c01r * ii + c01i * ir);
      const float i10r = -(c10r * ir - c10i * ii), i10i = -(c10r * ii + c10i * ir);
      const float i11r =  c00r * ir - c00i * ii, i11i =  c00r * ii + c00i * ir;
      const float* xf = &Xsh[f * XROW];
      const float x0r = xf[0], x0i = xf[1], x1r = xf[2], x1i = xf[3];
      #pragma unroll
      for (int s = 0; s < NS; ++s) {
        const float r00r = Rsh[s][0][0][0], r00i = Rsh[s][0][0][1];
        const float r01r = Rsh[s][0][1][0], r01i = Rsh[s][0][1][1];
        const float r10r = Rsh[s][1][0][0], r10i = Rsh[s][1][0][1];
        const float r11r = Rsh[s][1][1][0], r11i = Rsh[s][1][1][1];
        const float w = vs[s];
        // gain = w * R @ invCxx
        const float g00r = w * ((r00r * i00r - r00i * i00i) + (r01r * i10r - r01i * i10i));
        const float g00i = w * ((r00r * i00i + r00i * i00r) + (r01r * i10i + r01i * i10r));
        const float g01r = w * ((r00r * i01r - r00i * i01i) + (r01r * i11r - r01i * i11i));
        const float g01i = w * ((r00r * i01i + r00i * i01r) + (r01r * i11i + r01i * i11r));
        const float g10r = w * ((r10r * i00r - r10i * i00i) + (r11r * i10r - r11i * i10i));
        const float g10i = w * ((r10r * i00i + r10i * i00r) + (r11r * i10i + r11i * i10r));
        const float g11r = w * ((r10r * i01r - r10i * i01i) + (r11r * i11r - r11i * i11i));
        const float g11i = w * ((r10r * i01i + r10i * i01r) + (r11r * i11i + r11i * i11r));
        // y = gain @ x
        const float y0r = (g00r * x0r - g00i * x0i) + (g01r * x1r - g01i * x1i);
        const float y0i = (g00r * x0i + g00i * x0r) + (g01r * x1i + g01i * x1r);
        const float y1r = (g10r * x0r - g10i * x0i) + (g11r * x1r - g11i * x1i);
        const float y1i = (g10r * x0i + g10i * x0r) + (g11r * x1i + g11i * x1r);
        yf[s] = y0r; yf[4 + s] = y0i; yf[8 + s] = y1r; yf[12 + s] = y1i;
      }
    }
    __syncthreads();
  }

  // ---- write result tile: 2048 float4's, coalesced 64B per frame
  const size_t FSTR = (size_t)NB * YROW;
  #pragma unroll
  for (int i = 0; i < 8; ++i) {
    const int l = i * 256 + tid;      // float4 index within tile
    const int f = l >> 2;
    const int q = (l & 3) * 4;
    const float4 val = *(const float4*)&Ysh[f * YROW + q];
    *(float4*)&out[(size_t)f * FSTR + (size_t)b * YROW + q] = val;
  }
}

extern "C" void kernel_launch(void* const* d_in, const int* in_sizes, int n_in,
                              void* d_out, int out_size, void* d_ws, size_t ws_size,
                              hipStream_t stream) {
  (void)in_sizes; (void)n_in; (void)out_size; (void)d_ws; (void)ws_size;
  const float* y = (const float*)d_in[0];
  const float* x = (const float*)d_in[1];
  const int* iters = (const int*)d_in[2];
  float* out = (float*)d_out;
  em_wiener_kernel<<<dim3(NB), dim3(256), 0, stream>>>(y, x, iters, out);
}